// RNNModel_65833258713394
// MI455X (gfx1250) — compile-verified
//
#include <hip/hip_runtime.h>
#include <hip/hip_bf16.h>
#include <math.h>

typedef float v2f __attribute__((ext_vector_type(2)));
typedef float v8f __attribute__((ext_vector_type(8)));

#define ATN_EPS 1e-5f
#define NBLK 8            // workgroups in the persistent layer kernel

__device__ __forceinline__ float sigm(float x) { return 1.0f / (1.0f + expf(-x)); }

// Device-wide barrier: monotonic counter, NBLK blocks, one arrive per block.
// `want` = epoch * NBLK. Counter is zeroed host-side before each launch.
__device__ __forceinline__ void gbar(unsigned* cnt, unsigned want) {
  __syncthreads();
  if (threadIdx.x == 0) {
    __threadfence();                       // release stores
    atomicAdd(cnt, 1u);
    while (atomicAdd(cnt, 0u) < want) __builtin_amdgcn_s_sleep(2);
    __threadfence();                       // acquire
  }
  __syncthreads();
}

// ---------------------------------------------------------------------------
// C[M,N] = A[M,Kd] @ W[Kd,N] + bias[N]  (fp32, row-major).
// One wave -> one 16x64 strip via V_WMMA_F32_16X16X4_F32.
// Used for the hoisted x-path GEMM ([1024,512]@[512,2048]).
// ---------------------------------------------------------------------------
__global__ void gemm16_wmma(const float* __restrict__ A, const float* __restrict__ W,
                            const float* __restrict__ bias, float* __restrict__ C,
                            int M, int N, int Kd) {
  int gwave = (int)((blockIdx.x * blockDim.x + threadIdx.x) >> 5);
  int lane  = (int)(threadIdx.x & 31);
  int stripsN = N >> 6;
  int tileM = gwave / stripsN;
  int strip = gwave - tileM * stripsN;
  if (tileM * 16 >= M) return;                 // wave-uniform

  int half = lane >> 4;
  int lrow = lane & 15;
  const float* Arow = A + (size_t)(tileM * 16 + lrow) * Kd;
  int col0 = strip * 64 + lrow;

  v8f acc0 = {}; v8f acc1 = {}; v8f acc2 = {}; v8f acc3 = {};

  for (int k0 = 0; k0 < Kd; k0 += 4) {
    int ka = k0 + (half << 1);
    v2f a;
    a.x = Arow[ka];
    a.y = Arow[ka + 1];
    const float* w0 = W + (size_t)ka * N;
    const float* w1 = w0 + N;
    __builtin_prefetch(w0 + 4 * (size_t)N + col0, 0, 3);   // global_prefetch_b8
    v2f b0, b1, b2, b3;
    b0.x = w0[col0];      b0.y = w1[col0];
    b1.x = w0[col0 + 16]; b1.y = w1[col0 + 16];
    b2.x = w0[col0 + 32]; b2.y = w1[col0 + 32];
    b3.x = w0[col0 + 48]; b3.y = w1[col0 + 48];
    acc0 = __builtin_amdgcn_wmma_f32_16x16x4_f32(false, a, false, b0, (short)0, acc0, false, false);
    acc1 = __builtin_amdgcn_wmma_f32_16x16x4_f32(false, a, false, b1, (short)0, acc1, false, false);
    acc2 = __builtin_amdgcn_wmma_f32_16x16x4_f32(false, a, false, b2, (short)0, acc2, false, false);
    acc3 = __builtin_amdgcn_wmma_f32_16x16x4_f32(false, a, false, b3, (short)0, acc3, false, false);
  }

  float bv0 = bias[col0], bv1 = bias[col0 + 16], bv2 = bias[col0 + 32], bv3 = bias[col0 + 48];
#pragma unroll
  for (int r = 0; r < 8; ++r) {
    int row = tileM * 16 + r + (half << 3);
    float* crow = C + (size_t)row * N;
    crow[col0]      = acc0[r] + bv0;
    crow[col0 + 16] = acc1[r] + bv1;
    crow[col0 + 32] = acc2[r] + bv2;
    crow[col0 + 48] = acc3[r] + bv3;
  }
}

// ---------------------------------------------------------------------------
// Embedding gather: X[row,:] = emb[tokens[row],:]
// ---------------------------------------------------------------------------
__global__ void embed_gather(const int* __restrict__ tokens, const float* __restrict__ emb,
                             float* __restrict__ X) {
  int row = blockIdx.x;
  int tok = tokens[row];
  const float* src = emb + (size_t)tok * 512;
  float* dst = X + (size_t)row * 512;
  for (int i = threadIdx.x; i < 512; i += blockDim.x) dst[i] = src[i];
}

// ---------------------------------------------------------------------------
// Per-row sum / sumsq of [rows,2048] (wi window statistics, hoisted).
// ---------------------------------------------------------------------------
__global__ void rowsum2048(const float* __restrict__ A, float* __restrict__ S,
                           float* __restrict__ Q) {
  __shared__ float ss[256], qq[256];
  int row = blockIdx.x;
  const float* a = A + (size_t)row * 2048;
  float s = 0.f, q = 0.f;
  for (int i = threadIdx.x; i < 2048; i += 256) { float v = a[i]; s += v; q += v * v; }
  ss[threadIdx.x] = s; qq[threadIdx.x] = q;
  __syncthreads();
  for (int off = 128; off >= 1; off >>= 1) {
    if ((int)threadIdx.x < off) { ss[threadIdx.x] += ss[threadIdx.x + off];
                                  qq[threadIdx.x] += qq[threadIdx.x + off]; }
    __syncthreads();
  }
  if (threadIdx.x == 0) { S[row] = ss[0]; Q[row] = qq[0]; }
}

// ---------------------------------------------------------------------------
// Persistent per-layer scan kernel: NBLK=8 blocks x 256 threads (64 waves).
// Per step:
//   phase G: each wave computes one 16x32 strip of WH = h@w_hh + bh
//            (2 WMMA accumulator chains), stores it, and folds per-sample
//            sum/sumsq partials via __shfl_xor half-wave trees.
//   gbar
//   phase F: 128 threads per sample (2 samples per block): ATN stats
//            (wh from partials+ring, wi from hoisted Swi/Qwi), gates, c1,
//            block-local c stats + ring, c-norm, h1 -> h_state, Y.
//   gbar
// ---------------------------------------------------------------------------
__global__ __launch_bounds__(256)
void atn_lstm_layer(const float* __restrict__ Whh, const float* __restrict__ bh,
                    const float* __restrict__ WI,   // [T*16,2048], bias included
                    const float* __restrict__ Swi, const float* __restrict__ Qwi,
                    const float* __restrict__ aw_hh, const float* __restrict__ ab_hh,
                    const float* __restrict__ aw_ih, const float* __restrict__ ab_ih,
                    const float* __restrict__ aw_c, const float* __restrict__ ab_c,
                    float* __restrict__ WH,                     // [16,2048] scratch
                    float* __restrict__ Spart, float* __restrict__ Qpart, // [NBLK*16]
                    float* __restrict__ SwhR, float* __restrict__ QwhR,   // [5*16]
                    float* __restrict__ ScR, float* __restrict__ QcR,     // [5*16]
                    float* __restrict__ c_state, float* __restrict__ h_state,
                    float* __restrict__ Y,                       // [T,16,512]
                    unsigned* __restrict__ bar, int T) {
  __shared__ float wsum[8][16], wq[8][16];
  __shared__ float sred[256], qred[256];
  __shared__ float sMH[2], sRH[2], sMI[2], sRI[2], sMC[2], sRC[2];
  __shared__ float sSigO[1024];                // 2 samples x 512

  const int tid  = (int)threadIdx.x;
  const int wave = tid >> 5;
  const int lane = tid & 31;
  const int half = lane >> 4;
  const int lrow = lane & 15;
  const int strip = (int)blockIdx.x * 8 + wave;   // 0..63, one 32-col strip
  const int col0  = strip * 32 + lrow;
  const int bl = tid >> 7;                        // local sample 0..1 (fuse)
  const int gb = ((int)blockIdx.x << 1) + bl;     // sample 0..15 (fuse)
  const int ln = tid & 127;                       // lane within sample group
  unsigned epoch = 0;

  const float bv0 = bh[col0], bv1 = bh[col0 + 16];

  for (int t = 0; t < T; ++t) {
    // ------------------- phase G: WMMA strip GEMM ------------------------
    v8f acc0 = {}; v8f acc1 = {};
    const float* Arow = h_state + (size_t)lrow * 512;
    for (int k0 = 0; k0 < 512; k0 += 4) {
      int ka = k0 + (half << 1);
      v2f a;
      a.x = Arow[ka];
      a.y = Arow[ka + 1];
      const float* w0 = Whh + (size_t)ka * 2048;
      const float* w1 = w0 + 2048;
      __builtin_prefetch(w0 + 4 * 2048 + col0, 0, 3);
      v2f b0, b1;
      b0.x = w0[col0];      b0.y = w1[col0];
      b1.x = w0[col0 + 16]; b1.y = w1[col0 + 16];
      acc0 = __builtin_amdgcn_wmma_f32_16x16x4_f32(false, a, false, b0, (short)0, acc0, false, false);
      acc1 = __builtin_amdgcn_wmma_f32_16x16x4_f32(false, a, false, b1, (short)0, acc1, false, false);
    }

    // store strip + in-register per-sample partial stats
    float s8[8], q8[8];
#pragma unroll
    for (int r = 0; r < 8; ++r) {
      float v0 = acc0[r] + bv0, v1 = acc1[r] + bv1;
      int row = r + (half << 3);
      float* crow = WH + (size_t)row * 2048;
      crow[col0]      = v0;
      crow[col0 + 16] = v1;
      float s = v0 + v1;
      float q = v0 * v0 + v1 * v1;
#pragma unroll
      for (int m = 1; m < 16; m <<= 1) {       // reduce across 16-lane half
        s += __shfl_xor(s, m, 32);
        q += __shfl_xor(q, m, 32);
      }
      s8[r] = s; q8[r] = q;
    }
    if (lrow == 0) {
#pragma unroll
      for (int r = 0; r < 8; ++r) {
        wsum[wave][r + (half << 3)] = s8[r];
        wq[wave][r + (half << 3)]   = q8[r];
      }
    }
    __syncthreads();
    if (tid < 16) {
      float s = 0.f, q = 0.f;
#pragma unroll
      for (int w = 0; w < 8; ++w) { s += wsum[w][tid]; q += wq[w][tid]; }
      Spart[(int)blockIdx.x * 16 + tid] = s;
      Qpart[(int)blockIdx.x * 16 + tid] = q;
    }
    ++epoch;
    gbar(bar, epoch * NBLK);

    // ------------------- phase F: norms + gates + cell -------------------
    const int slot = t % 5;
    const float cntf = (float)((t + 1) < 5 ? (t + 1) : 5);
    if (ln == 0) {
      float sh = 0.f, qh = 0.f;
#pragma unroll
      for (int k = 0; k < NBLK; ++k) { sh += Spart[k * 16 + gb]; qh += Qpart[k * 16 + gb]; }
      SwhR[slot * 16 + gb] = sh; QwhR[slot * 16 + gb] = qh;
      float ssum = 0.f, qsum = 0.f;
      for (int j = 0; j < 5; ++j) {
        if (j == slot) { ssum += sh; qsum += qh; }
        else { ssum += SwhR[j * 16 + gb]; qsum += QwhR[j * 16 + gb]; }
      }
      float den = cntf * 2048.0f;
      float mu = ssum / den;
      float va = qsum / den - mu * mu;
      sMH[bl] = mu; sRH[bl] = rsqrtf(va + ATN_EPS);
      // wi window from hoisted per-step stats
      ssum = 0.f; qsum = 0.f;
      int j0 = (t - 4 > 0) ? (t - 4) : 0;
      for (int j = j0; j <= t; ++j) { ssum += Swi[j * 16 + gb]; qsum += Qwi[j * 16 + gb]; }
      mu = ssum / den;
      va = qsum / den - mu * mu;
      sMI[bl] = mu; sRI[bl] = rsqrtf(va + ATN_EPS);
    }
    __syncthreads();

    const float mh = sMH[bl], rh = sRH[bl], mi = sMI[bl], ri = sRI[bl];
    const float* WIt = WI + (size_t)t * 16 * 2048;
    float sc = 0.f, qc = 0.f;
    for (int m = 0; m < 4; ++m) {
      int j = ln + 128 * m;                // hidden index 0..511
      float z[4];
#pragma unroll
      for (int g4 = 0; g4 < 4; ++g4) {     // f,i,o,g quarters of 4H
        int gi = g4 * 512 + j;
        int idx = gb * 2048 + gi;
        float zh = (WH[idx] - mh) * rh * aw_hh[gi] + ab_hh[gi];
        float zi = (WIt[idx] - mi) * ri * aw_ih[gi] + ab_ih[gi];
        z[g4] = zh + zi;
      }
      float c_old = c_state[gb * 512 + j];
      float c1 = sigm(z[0]) * c_old + sigm(z[1]) * tanhf(z[3]);
      c_state[gb * 512 + j] = c1;
      sSigO[bl * 512 + j] = sigm(z[2]);
      sc += c1; qc += c1 * c1;
    }
    __syncthreads();
    sred[tid] = sc; qred[tid] = qc;
    __syncthreads();
    for (int off = 64; off >= 1; off >>= 1) {
      if (ln < off) { sred[tid] += sred[tid + off]; qred[tid] += qred[tid + off]; }
      __syncthreads();
    }
    if (ln == 0) {
      float ssc = sred[bl * 128], qqc = qred[bl * 128];
      ScR[slot * 16 + gb] = ssc; QcR[slot * 16 + gb] = qqc;
      float ssum = 0.f, qsum = 0.f;
      for (int j = 0; j < 5; ++j) {
        if (j == slot) { ssum += ssc; qsum += qqc; }
        else { ssum += ScR[j * 16 + gb]; qsum += QcR[j * 16 + gb]; }
      }
      float den = cntf * 512.0f;
      float mu = ssum / den;
      float va = qsum / den - mu * mu;
      sMC[bl] = mu; sRC[bl] = rsqrtf(va + ATN_EPS);
    }
    __syncthreads();

    const float mc = sMC[bl], rc = sRC[bl];
    float* yt = Y + (size_t)t * 16 * 512;
    for (int m = 0; m < 4; ++m) {
      int j = ln + 128 * m;
      float c1 = c_state[gb * 512 + j];
      float nc = (c1 - mc) * rc * aw_c[j] + ab_c[j];
      float h1 = sSigO[bl * 512 + j] * tanhf(nc);
      h_state[gb * 512 + j] = h1;
      yt[gb * 512 + j] = h1;
    }
    ++epoch;
    gbar(bar, epoch * NBLK);    // h_state visible to next step's GEMM
  }
}

// ---------------------------------------------------------------------------
// Host driver
// ---------------------------------------------------------------------------
extern "C" void kernel_launch(void* const* d_in, const int* in_sizes, int n_in,
                              void* d_out, int out_size, void* d_ws, size_t ws_size,
                              hipStream_t stream) {
  (void)in_sizes; (void)n_in; (void)out_size; (void)ws_size;
  const int* tokens = (const int*)d_in[0];
  const float* h0 = (const float*)d_in[1];
  const float* c0 = (const float*)d_in[2];
  const float* emb = (const float*)d_in[3];
  const float* P[2][10];
  for (int l = 0; l < 2; ++l)
    for (int p = 0; p < 10; ++p)
      P[l][p] = (const float*)d_in[4 + l * 10 + p];
  // P[l]: 0 w_ih, 1 w_hh, 2 bh, 3 bx, 4 aw_hh, 5 ab_hh, 6 aw_ih, 7 ab_ih, 8 aw_c, 9 ab_c

  float* ws = (float*)d_ws;
  size_t off = 0;
  float* X0   = ws + off; off += 1024 * 512;    // embedded inputs
  float* WIa  = ws + off; off += 1024 * 2048;   // x-path gates (reused per layer)
  float* Y0   = ws + off; off += 1024 * 512;    // layer-0 outputs
  float* WH   = ws + off; off += 16 * 2048;     // per-step h-path gates
  float* Swi  = ws + off; off += 1024;
  float* Qwi  = ws + off; off += 1024;
  float* RING = ws + off; off += 4 * 80;        // SwhR,QwhR,ScR,QcR
  float* SwhR = RING, *QwhR = RING + 80, *ScR = RING + 160, *QcR = RING + 240;
  float* Spart = ws + off; off += NBLK * 16;
  float* Qpart = ws + off; off += NBLK * 16;
  float* hS   = ws + off; off += 16 * 512;
  float* cS   = ws + off; off += 16 * 512;
  unsigned* bar = (unsigned*)(ws + off); off += 16;

  float* out = (float*)d_out;                   // result [1024,512]
  float* hn = out + 1024 * 512;                 // h_n [2,16,512]
  float* cn = hn + 2 * 16 * 512;                // c_n [2,16,512]

  embed_gather<<<1024, 256, 0, stream>>>(tokens, emb, X0);

  for (int l = 0; l < 2; ++l) {
    const float* Xin = (l == 0) ? X0 : Y0;

    // Hoisted x-path GEMM for all T: [1024,512] @ [512,2048] + bx
    {
      int tasks = (1024 / 16) * (2048 / 64);    // 2048 waves
      gemm16_wmma<<<(tasks * 32) / 256, 256, 0, stream>>>(
          Xin, P[l][0], P[l][3], WIa, 1024, 2048, 512);
    }
    rowsum2048<<<1024, 256, 0, stream>>>(WIa, Swi, Qwi);

    hipMemsetAsync(RING, 0, 4 * 80 * sizeof(float), stream);
    hipMemsetAsync(bar, 0, 16 * sizeof(float), stream);
    hipMemcpyAsync(hS, h0 + (size_t)l * 16 * 512, 16 * 512 * sizeof(float),
                   hipMemcpyDeviceToDevice, stream);
    hipMemcpyAsync(cS, c0 + (size_t)l * 16 * 512, 16 * 512 * sizeof(float),
                   hipMemcpyDeviceToDevice, stream);

    float* Ydst = (l == 0) ? Y0 : out;
    atn_lstm_layer<<<NBLK, 256, 0, stream>>>(
        P[l][1], P[l][2], WIa, Swi, Qwi,
        P[l][4], P[l][5], P[l][6], P[l][7], P[l][8], P[l][9],
        WH, Spart, Qpart, SwhR, QwhR, ScR, QcR,
        cS, hS, Ydst, bar, 64);

    hipMemcpyAsync(hn + (size_t)l * 16 * 512, hS, 16 * 512 * sizeof(float),
                   hipMemcpyDeviceToDevice, stream);
    hipMemcpyAsync(cn + (size_t)l * 16 * 512, cS, 16 * 512 * sizeof(float),
                   hipMemcpyDeviceToDevice, stream);
  }
}